// Mahalanobis_89541478187054
// MI455X (gfx1250) — compile-verified
//
#include <hip/hip_runtime.h>

typedef __attribute__((ext_vector_type(2))) float v2f;
typedef __attribute__((ext_vector_type(8))) float v8f;

#define N_    2048
#define D_    1024
#define S_    49
#define C_    1000
#define MPAD  3072   // 2048 + 1000 padded to tile multiple
#define CPAD  1024   // class dim padded for WMMA tiles
#define LDSP  144    // v2f row stride: 2*144 dwords % 64 == 32 -> disjoint bank halves

// ---------------------------------------------------------------------------
// 1) Pooling: z[n,d] = mean_s x[n,d,s], written into ZM rows [0,2048)
//    LDS-staged so global loads are fully coalesced (49 x 1KB wave loads).
__global__ __launch_bounds__(256) void pool_pack(const float* __restrict__ x,
                                                 float* __restrict__ ZM) {
    __shared__ float buf[256 * S_];           // 50,176 B
    const int t = threadIdx.x;
    const long base = (long)blockIdx.x * (256L * S_);
    for (int j = 0; j < S_; ++j) {
        int idx = j * 256 + t;                // coalesced across the block
        buf[idx] = x[base + idx];
    }
    __syncthreads();
    float s = 0.f;
#pragma unroll
    for (int j = 0; j < S_; ++j) s += buf[t * S_ + j];   // stride 49: conflict-free
    ZM[(long)blockIdx.x * 256 + t] = s * (1.0f / (float)S_);
}

// 2) ZM rows [2048, 3072): copy mu, zero the padding rows.
__global__ __launch_bounds__(256) void mu_pack(const float* __restrict__ mu,
                                               float* __restrict__ ZM) {
    int gid = blockIdx.x * 256 + threadIdx.x;   // 0 .. 1024*1024
    int r = gid >> 10;                          // 0..1023
    int d = gid & 1023;
    float v = (r < C_) ? mu[r * D_ + d] : 0.f;
    ZM[(long)(N_ + r) * D_ + d] = v;
}

// 3) Q = P + P^T ; muT[d,c] = mu[c,d] (zero-padded to CPAD columns).
__global__ __launch_bounds__(256) void prep_q_mut(const float* __restrict__ P,
                                                  const float* __restrict__ mu,
                                                  float* __restrict__ Q,
                                                  float* __restrict__ muT) {
    int gid = blockIdx.x * 256 + threadIdx.x;   // 1M
    int i = gid >> 10;
    int j = gid & 1023;
    Q[gid]  = P[i * D_ + j] + P[j * D_ + i];
    muT[gid] = (j < C_) ? mu[j * D_ + i] : 0.f;
}

// 4) fp32 WMMA GEMM: C[M,N] = A[M,K] * B[K,N], row-major, M,N multiples of
//    128, K multiple of 16. Block 256 threads = 8 waves (2x4), each wave owns
//    a 64x32 tile (4x2 accumulators). LDS holds K-pair-interleaved v2f tiles:
//    each WMMA fragment is ONE aligned ds_load_b64 -> even VGPR pair, no movs.
//    Double-buffered: one barrier per 16-wide K stage.
__global__ __launch_bounds__(256) void gemm_wmma_f32(
    const float* __restrict__ A, const float* __restrict__ B,
    float* __restrict__ C, int M, int N, int K, int lda, int ldb, int ldc) {
    __shared__ v2f sA[2][8 * LDSP];   // [buf][k/2][m]: {A[m][k], A[m][k+1]}
    __shared__ v2f sB[2][8 * LDSP];   // [buf][k/2][n]: {B[k][n], B[k+1][n]}

    const int t     = threadIdx.x;
    const int lane  = t & 31;
    const int wave  = t >> 5;
    const int wm    = wave >> 2;      // 0..1  -> 64-row slab
    const int wn    = wave & 3;       // 0..3  -> 32-col slab
    const int lhalf = lane >> 4;      // 0/1
    const int l16   = lane & 15;
    const int M0 = blockIdx.y * 128;
    const int N0 = blockIdx.x * 128;

    // staging assignments
    const int rowA0 = t >> 2;         // 0..63
    const int rowA1 = 64 + rowA0;     // 64..127
    const int cpA   = t & 3;          // float4 column quad within 16-wide K
    const int kpB   = t >> 5;         // 0..7 K-pair for B staging
    const int cqB   = t & 31;         // 0..31 column quad for B staging

    const float* Ag0 = A + (long)(M0 + rowA0) * lda + cpA * 4;
    const float* Ag1 = A + (long)(M0 + rowA1) * lda + cpA * 4;
    const float* Bg0 = B + (long)(2 * kpB) * ldb + N0 + cqB * 4;
    const float* Bg1 = Bg0 + ldb;

    v8f acc[4][2];
#pragma unroll
    for (int im = 0; im < 4; ++im)
#pragma unroll
        for (int in2 = 0; in2 < 2; ++in2)
#pragma unroll
            for (int v = 0; v < 8; ++v) acc[im][in2][v] = 0.f;

    const int nstages = K / 16;

    // prologue: stage 0 -> buffer 0
    {
        const float4 a0 = *(const float4*)(Ag0);
        const float4 a1 = *(const float4*)(Ag1);
        const float4 b0 = *(const float4*)(Bg0);
        const float4 b1 = *(const float4*)(Bg1);
        sA[0][(cpA * 2 + 0) * LDSP + rowA0] = (v2f){a0.x, a0.y};
        sA[0][(cpA * 2 + 1) * LDSP + rowA0] = (v2f){a0.z, a0.w};
        sA[0][(cpA * 2 + 0) * LDSP + rowA1] = (v2f){a1.x, a1.y};
        sA[0][(cpA * 2 + 1) * LDSP + rowA1] = (v2f){a1.z, a1.w};
        sB[0][kpB * LDSP + cqB * 4 + 0] = (v2f){b0.x, b1.x};
        sB[0][kpB * LDSP + cqB * 4 + 1] = (v2f){b0.y, b1.y};
        sB[0][kpB * LDSP + cqB * 4 + 2] = (v2f){b0.z, b1.z};
        sB[0][kpB * LDSP + cqB * 4 + 3] = (v2f){b0.w, b1.w};
    }
    __syncthreads();

    for (int s = 0; s < nstages; ++s) {
        const int buf = s & 1;
        const bool more = (s + 1) < nstages;

        // issue next stage's global loads early (latency hidden behind WMMA)
        float4 a0, a1, b0, b1;
        if (more) {
            const int k1 = (s + 1) * 16;
            a0 = *(const float4*)(Ag0 + k1);
            a1 = *(const float4*)(Ag1 + k1);
            b0 = *(const float4*)(Bg0 + (long)k1 * ldb);
            b1 = *(const float4*)(Bg1 + (long)k1 * ldb);
        }

#pragma unroll
        for (int kk = 0; kk < 4; ++kk) {
            // fp32 16x16x4 layout: lanes0-15 K=kb,kb+1; lanes16-31 K=kb+2,kb+3
            const int kp = kk * 2 + lhalf;      // K-pair plane for this half-wave
            v2f af[4], bf[2];
#pragma unroll
            for (int im = 0; im < 4; ++im)
                af[im] = sA[buf][kp * LDSP + wm * 64 + im * 16 + l16];
#pragma unroll
            for (int in2 = 0; in2 < 2; ++in2)
                bf[in2] = sB[buf][kp * LDSP + wn * 32 + in2 * 16 + l16];
#pragma unroll
            for (int im = 0; im < 4; ++im)
#pragma unroll
                for (int in2 = 0; in2 < 2; ++in2)
                    acc[im][in2] = __builtin_amdgcn_wmma_f32_16x16x4_f32(
                        false, af[im], false, bf[in2], (short)0, acc[im][in2],
                        false, false);
        }

        if (more) {
            const int nb = buf ^ 1;
            sA[nb][(cpA * 2 + 0) * LDSP + rowA0] = (v2f){a0.x, a0.y};
            sA[nb][(cpA * 2 + 1) * LDSP + rowA0] = (v2f){a0.z, a0.w};
            sA[nb][(cpA * 2 + 0) * LDSP + rowA1] = (v2f){a1.x, a1.y};
            sA[nb][(cpA * 2 + 1) * LDSP + rowA1] = (v2f){a1.z, a1.w};
            sB[nb][kpB * LDSP + cqB * 4 + 0] = (v2f){b0.x, b1.x};
            sB[nb][kpB * LDSP + cqB * 4 + 1] = (v2f){b0.y, b1.y};
            sB[nb][kpB * LDSP + cqB * 4 + 2] = (v2f){b0.z, b1.z};
            sB[nb][kpB * LDSP + cqB * 4 + 3] = (v2f){b0.w, b1.w};
        }
        __syncthreads();
    }

    // Epilogue: C/D layout — VGPR v: row = base + v + 8*lhalf, col = l16
#pragma unroll
    for (int im = 0; im < 4; ++im)
#pragma unroll
        for (int in2 = 0; in2 < 2; ++in2)
#pragma unroll
            for (int v = 0; v < 8; ++v) {
                int row = M0 + wm * 64 + im * 16 + v + 8 * lhalf;
                int col = N0 + wn * 32 + in2 * 16 + l16;
                C[(long)row * ldc + col] = acc[im][in2][v];
            }
}

// 5) rbuf[r] = dot(F[r,:], ZM[r,:]) for r in [0, 3048)
__global__ __launch_bounds__(256) void rowdot(const float* __restrict__ F,
                                              const float* __restrict__ ZM,
                                              float* __restrict__ rbuf) {
    __shared__ float red[256];
    const int r = blockIdx.x;
    const int t = threadIdx.x;
    const float4 a = *(const float4*)(F + (long)r * D_ + t * 4);
    const float4 b = *(const float4*)(ZM + (long)r * D_ + t * 4);
    red[t] = a.x * b.x + a.y * b.y + a.z * b.z + a.w * b.w;
    __syncthreads();
    for (int off = 128; off > 0; off >>= 1) {
        if (t < off) red[t] += red[t + off];
        __syncthreads();
    }
    if (t == 0) rbuf[r] = red[0];
}

// 6) score[n,c] = 0.5*CR[n,c] - 0.25*(rbuf[n] + rbuf[2048+c])
__global__ __launch_bounds__(256) void finalize(const float* __restrict__ CR,
                                                const float* __restrict__ rbuf,
                                                float* __restrict__ out) {
    long gid = (long)blockIdx.x * 256 + threadIdx.x;
    if (gid >= (long)N_ * C_) return;
    int n = (int)(gid / C_);
    int c = (int)(gid % C_);
    out[gid] = 0.5f * CR[(long)n * CPAD + c] - 0.25f * (rbuf[n] + rbuf[N_ + c]);
}

extern "C" void kernel_launch(void* const* d_in, const int* in_sizes, int n_in,
                              void* d_out, int out_size, void* d_ws, size_t ws_size,
                              hipStream_t stream) {
    const float* x  = (const float*)d_in[0];   // [2048,1024,49]
    const float* mu = (const float*)d_in[1];   // [1000,1024]
    const float* P  = (const float*)d_in[2];   // [1024,1024]
    float* out = (float*)d_out;                // [2048,1000]

    float* ws  = (float*)d_ws;
    float* ZM  = ws;                              // [3072,1024] = [z ; mu ; 0]
    float* F   = ZM  + (size_t)MPAD * D_;         // [3072,1024] = ZM @ Q
    float* Q   = F   + (size_t)MPAD * D_;         // [1024,1024] = P + P^T
    float* muT = Q   + (size_t)D_ * D_;           // [1024,1024] (zero-padded)
    float* CR  = muT + (size_t)D_ * D_;           // [2048,1024] = F[0:2048] @ muT
    float* rb  = CR  + (size_t)N_ * CPAD;         // [3072] row dots

    pool_pack <<<(N_ * D_) / 256, 256, 0, stream>>>(x, ZM);
    mu_pack   <<<(CPAD * D_) / 256, 256, 0, stream>>>(mu, ZM);
    prep_q_mut<<<(D_ * D_) / 256, 256, 0, stream>>>(P, mu, Q, muT);

    // F = ZM @ Q : M=3072, N=1024, K=1024
    gemm_wmma_f32<<<dim3(1024 / 128, MPAD / 128), 256, 0, stream>>>(
        ZM, Q, F, MPAD, 1024, 1024, D_, D_, D_);

    rowdot<<<N_ + C_, 256, 0, stream>>>(F, ZM, rb);

    // CR = F[0:2048] @ muT : M=2048, N=1024, K=1024
    gemm_wmma_f32<<<dim3(CPAD / 128, N_ / 128), 256, 0, stream>>>(
        F, muT, CR, N_, CPAD, 1024, D_, D_, CPAD);

    finalize<<<(N_ * C_) / 256, 256, 0, stream>>>(CR, rb, out);
}